// GConv_5686536700488
// MI455X (gfx1250) — compile-verified
//
#include <hip/hip_runtime.h>
#include <cstddef>
#include <cstdint>

// ---------------------------------------------------------------------------
// GCN-style relational conv, K=4 rounds, D=512, fp32 end-to-end.
//  - GEMMs: CDNA5 V_WMMA_F32_16X16X4_F32 (fp32 in, fp32 accum). Compute is far
//    from the roofline limit (scatter atomics dominate; x fits in 192MB L2),
//    so fp32 WMMA keeps reference numerics at no real cost.
//  - A-tiles staged global->LDS with GLOBAL_LOAD_ASYNC_TO_LDS_B128 (ASYNCcnt),
//    double-buffered so chunk k+1 loads overlap chunk k WMMA compute.
//  - Scatter-add uses native no-return global_atomic_add_f32.
// ---------------------------------------------------------------------------

typedef __attribute__((ext_vector_type(2))) float v2f;
typedef __attribute__((ext_vector_type(8))) float v8f;

#define DIM        512
#define KCHUNK     128
#define NCHUNK     (DIM / KCHUNK)        // 4
#define LDSSTRIDE  132   // 128 + 4 pad: (132 mod 64)=4 -> conflict-free column
                         // reads; multiple of 4 floats -> 16B-aligned rows

// ---- transpose 512x512 weight so B-fragment loads are lane-coalesced -------
__global__ void transpose_k(const float* __restrict__ W, float* __restrict__ WT) {
    int idx = blockIdx.x * blockDim.x + threadIdx.x;   // DIM*DIM threads
    int k = idx & (DIM - 1);
    int j = idx >> 9;
    WT[(size_t)k * DIM + j] = W[(size_t)j * DIM + k];
}

// ---- agg[dst] += x[src], one block per edge, float4 per thread -------------
__global__ __launch_bounds__(DIM / 4)
void scatter_k(const float* __restrict__ x,
               const long long* __restrict__ src,
               const long long* __restrict__ dst,
               float* __restrict__ agg) {
    const int e = blockIdx.x;
    const int c = threadIdx.x << 2;
    const long long s = src[e];
    const long long t = dst[e];
    const float4 v = *(const float4*)(x + (size_t)s * DIM + c);   // coalesced 2KB/row
    float* a = agg + (size_t)t * DIM + c;
    // relaxed agent-scope fp32 atomics -> native global_atomic_add_f32 (no-return)
    __hip_atomic_fetch_add(a + 0, v.x, __ATOMIC_RELAXED, __HIP_MEMORY_SCOPE_AGENT);
    __hip_atomic_fetch_add(a + 1, v.y, __ATOMIC_RELAXED, __HIP_MEMORY_SCOPE_AGENT);
    __hip_atomic_fetch_add(a + 2, v.z, __ATOMIC_RELAXED, __HIP_MEMORY_SCOPE_AGENT);
    __hip_atomic_fetch_add(a + 3, v.w, __ATOMIC_RELAXED, __HIP_MEMORY_SCOPE_AGENT);
}

// Issue async global->LDS B128 copies for one 16 x KCHUNK chunk of agg and x.
// VDST operand = 32-bit LDS byte address (low 32 bits of the generic pointer,
// per the CDNA5 LDS aperture rule); VADDR = 64-bit global address.
__device__ __forceinline__ void stage_chunk_async(
    float* __restrict__ dA, float* __restrict__ dX,
    const float* __restrict__ agg, const float* __restrict__ xc,
    int r0, int kc, int tid) {
#pragma unroll
    for (int i = tid; i < 16 * (KCHUNK / 4); i += 128) {   // 4 iters/thread
        const int row = i >> 5;               // KCHUNK/4 = 32 float4 per row
        const int c4  = (i & 31) << 2;
        const unsigned la = (unsigned)(uintptr_t)(&dA[row * LDSSTRIDE + c4]);
        const unsigned lx = (unsigned)(uintptr_t)(&dX[row * LDSSTRIDE + c4]);
        const float* ga = &agg[(size_t)(r0 + row) * DIM + kc + c4];
        const float* gx = &xc[(size_t)(r0 + row) * DIM + kc + c4];
        asm volatile("global_load_async_to_lds_b128 %0, %1, off"
                     :: "v"(la), "v"(ga) : "memory");
        asm volatile("global_load_async_to_lds_b128 %0, %1, off"
                     :: "v"(lx), "v"(gx) : "memory");
    }
}

__device__ __forceinline__ void wait_async_zero() {
    asm volatile("s_wait_asynccnt 0x0" ::: "memory");
}

// ---- y = relu(agg @ Wrel^T + x @ Wroot^T), fused dual GEMM -----------------
// Block: 128 threads = 4 waves. Block tile: 16 rows x 64 cols of y.
// Wave w owns cols [blockIdx.y*64 + w*16, +16). K double-buffered in LDS.
__global__ __launch_bounds__(128)
void gemm_relu_k(const float* __restrict__ agg,
                 const float* __restrict__ xc,
                 const float* __restrict__ WrelT,
                 const float* __restrict__ WrootT,
                 float* __restrict__ y) {
    __shared__ float ldsA[2][16 * LDSSTRIDE];
    __shared__ float ldsX[2][16 * LDSSTRIDE];

    const int tid  = threadIdx.x;
    const int lane = tid & 31;
    const int wave = tid >> 5;
    const int r0   = blockIdx.x * 16;
    const int colb = blockIdx.y * 64 + wave * 16;

    // WMMA f32 16x16x4 per-lane fragment indices:
    //  A (16x4):  lanes 0-15 -> M=lane, K={0,1}; lanes 16-31 -> M=lane-16, K={2,3}
    //  B (4x16):  mirrored:   n = lane&15, K pair selected by lane>>4
    //  C/D (16x16): vgpr r -> M = r + 8*(lane>>4), N = lane&15
    const int m  = lane & 15;
    const int kh = lane >> 4;

    v8f acc_rel  = {0.f, 0.f, 0.f, 0.f, 0.f, 0.f, 0.f, 0.f};
    v8f acc_root = {0.f, 0.f, 0.f, 0.f, 0.f, 0.f, 0.f, 0.f};

    // prologue: stage chunk 0
    stage_chunk_async(ldsA[0], ldsX[0], agg, xc, r0, 0, tid);
    wait_async_zero();
    __syncthreads();

    for (int c = 0; c < NCHUNK; ++c) {
        const int cur = c & 1;
        // overlap: issue chunk c+1 async copies before computing chunk c
        if (c + 1 < NCHUNK)
            stage_chunk_async(ldsA[cur ^ 1], ldsX[cur ^ 1], agg, xc,
                              r0, (c + 1) * KCHUNK, tid);

        const float* bufA = ldsA[cur];
        const float* bufX = ldsX[cur];
        const int kc = c * KCHUNK;

#pragma unroll 4
        for (int k0 = 0; k0 < KCHUNK; k0 += 4) {
            const int kb = k0 + 2 * kh;     // this lane's K pair within chunk
            const int kg = kc + kb;         // global K for weight fetch

            v2f a1; a1.x = bufA[m * LDSSTRIDE + kb];
                    a1.y = bufA[m * LDSSTRIDE + kb + 1];
            v2f b1; b1.x = WrelT[(size_t)(kg + 0) * DIM + colb + m];
                    b1.y = WrelT[(size_t)(kg + 1) * DIM + colb + m];
            acc_rel = __builtin_amdgcn_wmma_f32_16x16x4_f32(
                false, a1, false, b1, (short)0, acc_rel, false, false);

            v2f a2; a2.x = bufX[m * LDSSTRIDE + kb];
                    a2.y = bufX[m * LDSSTRIDE + kb + 1];
            v2f b2; b2.x = WrootT[(size_t)(kg + 0) * DIM + colb + m];
                    b2.y = WrootT[(size_t)(kg + 1) * DIM + colb + m];
            acc_root = __builtin_amdgcn_wmma_f32_16x16x4_f32(
                false, a2, false, b2, (short)0, acc_root, false, false);
        }

        // chunk c+1 LDS writes complete; all waves done reading chunk c
        wait_async_zero();
        __syncthreads();
    }

    // combine, relu, store (each half-wave writes 16 consecutive floats)
#pragma unroll
    for (int r = 0; r < 8; ++r) {
        float v = acc_rel[r] + acc_root[r];
        v = v > 0.f ? v : 0.f;
        y[(size_t)(r0 + r + 8 * kh) * DIM + colb + m] = v;
    }
}

// ---- out[n] = sum of columns 15, 31, ..., 511 ------------------------------
__global__ void colsum_k(const float* __restrict__ x, float* __restrict__ out, int n) {
    int i = blockIdx.x * blockDim.x + threadIdx.x;
    if (i < n) {
        float s = 0.f;
        const float* row = x + (size_t)i * DIM;
#pragma unroll
        for (int j = 15; j < DIM; j += 16) s += row[j];
        out[i] = s;
    }
}

extern "C" void kernel_launch(void* const* d_in, const int* in_sizes, int n_in,
                              void* d_out, int out_size, void* d_ws, size_t ws_size,
                              hipStream_t stream) {
    const float*     x     = (const float*)d_in[0];
    const float*     Wrel  = (const float*)d_in[1];
    const float*     Wroot = (const float*)d_in[2];
    const long long* ei    = (const long long*)d_in[3];   // int64 (2, E)
    // d_in[4] (batch) unused by the reference math

    const int n_nodes = in_sizes[0] / DIM;   // 20000 (divisible by 16)
    const int n_edges = in_sizes[3] / 2;     // 160000
    const long long* src = ei;
    const long long* dst = ei + n_edges;

    const size_t wbytes = (size_t)DIM * DIM * sizeof(float);     // 1 MB
    const size_t xbytes = (size_t)n_nodes * DIM * sizeof(float); // ~41 MB
    char* ws = (char*)d_ws;
    float* WrelT  = (float*)(ws);
    float* WrootT = (float*)(ws + wbytes);
    float* agg    = (float*)(ws + 2 * wbytes);
    float* b0     = (float*)(ws + 2 * wbytes + xbytes);
    float* b1     = (float*)(ws + 2 * wbytes + 2 * xbytes);

    transpose_k<<<(DIM * DIM) / 256, 256, 0, stream>>>(Wrel,  WrelT);
    transpose_k<<<(DIM * DIM) / 256, 256, 0, stream>>>(Wroot, WrootT);

    const float* cur = x;
    float* nxt = b0;
    for (int it = 0; it < 4; ++it) {
        hipMemsetAsync(agg, 0, xbytes, stream);
        scatter_k<<<n_edges, DIM / 4, 0, stream>>>(cur, src, dst, agg);
        dim3 grid(n_nodes / 16, DIM / 64);
        gemm_relu_k<<<grid, 128, 0, stream>>>(agg, cur, WrelT, WrootT, nxt);
        cur = nxt;
        nxt = (nxt == b0) ? b1 : b0;
    }

    colsum_k<<<(n_nodes + 255) / 256, 256, 0, stream>>>(cur, (float*)d_out, n_nodes);
}